// Vector3DFeaturesLatent_50457275793731
// MI455X (gfx1250) — compile-verified
//
#include <hip/hip_runtime.h>
#include <hip/hip_bf16.h>

typedef __attribute__((ext_vector_type(2))) float v2f;
typedef __attribute__((ext_vector_type(8))) float v8f;

#define Jv    19
#define Pv    100000
#define Rv    16
#define NINv  32
#define NOUTv 36
#define Fv    12
#define FEATv (NOUTv * Rv)          // 576 floats per volume
#define NPAD  48                    // N padded to 3 WMMA tiles
#define WP_OFF  (Jv * FEATv)        // ws offset of W' [48][36]
#define B_OFF   (WP_OFF + NPAD * NOUTv) // ws offset of padded bias [48]
#define PREP_TOTAL (B_OFF + NPAD)

// ---------------------------------------------------------------------------
// Prep kernel (tiny, one launch):
//  (a) feat[j][o] = latent[j,:] . pl_w[j,:,o] + pl_b[j][o], written in the
//      K'-ordering layout [j][c][f][r]  (k' = c*12 + f, addr = k'*16 + r)
//  (b) W'[n][k'] = map_w[n][f*3+c]  (K permuted to axis-major), rows 36..47 = 0
//  (c) padded bias[48]
// ---------------------------------------------------------------------------
__global__ void prep_kernel(const float* __restrict__ latent,
                            const float* __restrict__ pl_w,
                            const float* __restrict__ pl_b,
                            const float* __restrict__ map_w,
                            const float* __restrict__ map_b,
                            float* __restrict__ ws) {
    int g = blockIdx.x * blockDim.x + threadIdx.x;
    if (g < WP_OFF) {
        int j = g / FEATv;
        int o = g % FEATv;                 // o = f*(R*3) + r*3 + c (reference)
        int f = o / (Rv * 3);
        int r = (o / 3) % Rv;
        int c = o % 3;
        float acc = pl_b[j * FEATv + o];
        const float* lj = latent + j * NINv;
        const float* wj = pl_w + (size_t)j * NINv * FEATv + o;
#pragma unroll
        for (int i = 0; i < NINv; ++i)
            acc += lj[i] * wj[(size_t)i * FEATv];
        ws[j * FEATv + (c * Fv + f) * Rv + r] = acc;     // [j][c][f][r]
    } else if (g < B_OFF) {
        int t  = g - WP_OFF;
        int n  = t / NOUTv;
        int kp = t % NOUTv;                // k' = c*12 + f
        int c  = kp / Fv;
        int f  = kp % Fv;
        ws[g] = (n < NOUTv) ? map_w[n * NOUTv + f * 3 + c] : 0.0f;
    } else if (g < PREP_TOTAL) {
        int n = g - B_OFF;
        ws[g] = (n < NOUTv) ? map_b[n] : 0.0f;
    }
}

// ---------------------------------------------------------------------------
// Fused sampler + 36x36 mapper. One wave = 4 M-tiles (64 points) of one j.
// With k' = c*12+f, each K-chunk of 4 lies in one axis: c = kk/3 is a
// compile-time constant per unrolled chunk -> no runtime axis selection.
// 108 v_wmma_f32_16x16x4_f32 per wave. Non-temporal streaming stores.
// ---------------------------------------------------------------------------
__global__ void __launch_bounds__(256)
fused_sample_map_kernel(const float* __restrict__ pts,
                        const float* __restrict__ ws,
                        float* __restrict__ out) {
    const float* feat_all = ws;
    const float* wprime   = ws + WP_OFF;    // [48][36]
    const float* biasp    = ws + B_OFF;     // [48]

    __shared__ float featL[FEATv];          // [c][f][r] for this j
    const int j = blockIdx.y;
    for (int idx = threadIdx.x; idx < FEATv; idx += 256)
        featL[idx] = feat_all[j * FEATv + idx];
    __syncthreads();

    const int lane = threadIdx.x & 31;
    const int wave = threadIdx.x >> 5;
    const int half = lane >> 4;             // 0: K%4 in {0,1}; 1: {2,3}
    const int lm   = lane & 15;
    const int p0   = (blockIdx.x * 8 + wave) * 64;   // 4 tiles of 16 points
    const int hoff = 32 * half;             // (2*half)*16 element offset

    // ---- B fragments: aligned b64 loads from reordered W' (wave-invariant) ----
    v2f   bfrag[3][9];
    float bias[3];
#pragma unroll
    for (int nt = 0; nt < 3; ++nt) {
        const float* row = wprime + (nt * 16 + lm) * NOUTv + 2 * half;
#pragma unroll
        for (int kk = 0; kk < 9; ++kk)
            bfrag[nt][kk] = *(const v2f*)(row + kk * 4);
        bias[nt] = biasp[nt * 16 + lm];
    }

    // ---- per-tile interpolation state: LDS base index + weight per axis ----
    int   foff[4][3];
    float wsel[4][3];
#pragma unroll
    for (int m = 0; m < 4; ++m) {
        int p = p0 + m * 16 + lm;
        if (p >= Pv) p = Pv - 1;            // clamp; stores are guarded
        const float* pp = pts + ((size_t)j * Pv + p) * 3;
#pragma unroll
        for (int c = 0; c < 3; ++c) {
            float u  = (pp[c] + 1.0f) * (0.5f * (float)(Rv - 1));
            u        = fminf(fmaxf(u, 0.0f), (float)(Rv - 1));
            float fi = fminf(floorf(u), (float)(Rv - 2));
            foff[m][c] = c * (Fv * Rv) + (int)fi;   // featL index of r=i0
            wsel[m][c] = u - fi;
        }
    }

    // ---- accumulators seeded with bias ----
    v8f acc[4][3];
#pragma unroll
    for (int m = 0; m < 4; ++m)
#pragma unroll
        for (int nt = 0; nt < 3; ++nt)
#pragma unroll
            for (int v = 0; v < 8; ++v) acc[m][nt][v] = bias[nt];

    // ---- K loop: 9 chunks x 4 tiles x 3 N-tiles = 108 WMMAs ----
#pragma unroll
    for (int kk = 0; kk < 9; ++kk) {
        const int c     = kk / 3;           // compile-time per unrolled chunk
        const int fbase = (kk % 3) * 4;     // + 2*half + t
#pragma unroll
        for (int m = 0; m < 4; ++m) {
            const int base = foff[m][c] + fbase * Rv + hoff;
            const float w  = wsel[m][c];
            v2f a;
#pragma unroll
            for (int t = 0; t < 2; ++t) {
                float g0 = featL[base + t * Rv];
                float g1 = featL[base + t * Rv + 1];
                a[t] = g0 + (g1 - g0) * w;
            }
#pragma unroll
            for (int nt = 0; nt < 3; ++nt) {
                acc[m][nt] = __builtin_amdgcn_wmma_f32_16x16x4_f32(
                    false, a, false, bfrag[nt][kk], (short)0,
                    acc[m][nt], false, false);
            }
        }
    }

    // ---- streaming stores: lane%16 -> channel, VGPR v -> point row ----
#pragma unroll
    for (int m = 0; m < 4; ++m) {
        const int prow0 = p0 + m * 16 + 8 * half;
#pragma unroll
        for (int nt = 0; nt < 3; ++nt) {
            int n = nt * 16 + lm;
            if (n >= NOUTv) continue;
#pragma unroll
            for (int v = 0; v < 8; ++v) {
                int prow = prow0 + v;
                if (prow < Pv)
                    __builtin_nontemporal_store(acc[m][nt][v],
                        &out[((size_t)j * Pv + prow) * NOUTv + n]);
            }
        }
    }
}

// ---------------------------------------------------------------------------
extern "C" void kernel_launch(void* const* d_in, const int* in_sizes, int n_in,
                              void* d_out, int out_size, void* d_ws, size_t ws_size,
                              hipStream_t stream) {
    const float* pts    = (const float*)d_in[0];
    const float* latent = (const float*)d_in[1];
    const float* pl_w   = (const float*)d_in[2];
    const float* pl_b   = (const float*)d_in[3];
    const float* map_w  = (const float*)d_in[4];
    const float* map_b  = (const float*)d_in[5];
    float* out = (float*)d_out;
    float* ws  = (float*)d_ws;   // PREP_TOTAL floats ~= 51 KB

    prep_kernel<<<dim3((PREP_TOTAL + 255) / 256), 256, 0, stream>>>(
        latent, pl_w, pl_b, map_w, map_b, ws);

    dim3 grid((Pv + 511) / 512, Jv);     // 8 waves/block x 64 pts/wave
    fused_sample_map_kernel<<<grid, 256, 0, stream>>>(pts, ws, out);
}